// PooledTransformerDecoder_37649683317426
// MI455X (gfx1250) — compile-verified
//
#include <hip/hip_runtime.h>
#include <math.h>

// Model constants from the reference
#define VOCAB 32000
#define DMODEL 512
#define NHEAD 8
#define HDIM 64
#define HFF 2048
#define NLAYER 2
#define BATCH 8
#define LMAX 48
#define SMAX 49   // LMAX + 1

typedef float v2f __attribute__((ext_vector_type(2)));
typedef float v8f __attribute__((ext_vector_type(8)));

// ---------------------------------------------------------------------------
// prep: init token buffer (SOS=1 at position 0) and zero padded h buffer
// ---------------------------------------------------------------------------
__global__ void prep_init_kernel(float* hbuf, int* tok) {
    int tid = threadIdx.x;
    for (int i = tid; i < 16 * DMODEL; i += 256) hbuf[i] = 0.0f;
    for (int i = tid; i < BATCH * SMAX; i += 256) tok[i] = (i % SMAX == 0) ? 1 : 0;
}

// ---------------------------------------------------------------------------
// prep: cross-attention output is constant per (layer, batch):
//   v = memory[b] @ wv_ca^T + bv_ca ;  ca_add = v @ ca_out_w^T + ca_out_b
// (softmax over a single key == 1 for every query/head)
// ---------------------------------------------------------------------------
__global__ void prep_ca_kernel(const float* __restrict__ memory,
                               const float* __restrict__ ca_in_w,
                               const float* __restrict__ ca_in_b,
                               const float* __restrict__ ca_out_w,
                               const float* __restrict__ ca_out_b,
                               float* __restrict__ ca_add) {
    __shared__ float mm[DMODEL];
    __shared__ float vv[DMODEL];
    int l = blockIdx.x / BATCH;
    int b = blockIdx.x % BATCH;
    int tid = threadIdx.x;
    for (int i = tid; i < DMODEL; i += 256) mm[i] = memory[b * DMODEL + i];
    __syncthreads();
    // wv rows are [2D, 3D) of ca_in_w[l]
    for (int i = tid; i < DMODEL; i += 256) {
        float acc = ca_in_b[l * 3 * DMODEL + 2 * DMODEL + i];
        const float* row = ca_in_w + (size_t)(l * 3 * DMODEL + 2 * DMODEL + i) * DMODEL;
        for (int k = 0; k < DMODEL; k++) acc += mm[k] * row[k];
        vv[i] = acc;
    }
    __syncthreads();
    for (int i = tid; i < DMODEL; i += 256) {
        float acc = ca_out_b[l * DMODEL + i];
        const float* row = ca_out_w + (size_t)(l * DMODEL + i) * DMODEL;
        for (int k = 0; k < DMODEL; k++) acc += vv[k] * row[k];
        ca_add[(l * BATCH + b) * DMODEL + i] = acc;
    }
}

// ---------------------------------------------------------------------------
// Block-wide LayerNorm over 512 elements (256 threads)
// ---------------------------------------------------------------------------
__device__ __forceinline__ void block_ln(float* dst, const float* src,
                                         const float* g, const float* bb,
                                         float* red) {
    int tid = threadIdx.x;
    float s = 0.0f;
    for (int i = tid; i < DMODEL; i += 256) s += src[i];
    red[tid] = s;
    __syncthreads();
    for (int st = 128; st > 0; st >>= 1) {
        if (tid < st) red[tid] += red[tid + st];
        __syncthreads();
    }
    float mean = red[0] * (1.0f / DMODEL);
    __syncthreads();
    float vs = 0.0f;
    for (int i = tid; i < DMODEL; i += 256) {
        float d = src[i] - mean;
        vs += d * d;
    }
    red[tid] = vs;
    __syncthreads();
    for (int st = 128; st > 0; st >>= 1) {
        if (tid < st) red[tid] += red[tid + st];
        __syncthreads();
    }
    float rstd = rsqrtf(red[0] * (1.0f / DMODEL) + 1e-5f);
    __syncthreads();
    for (int i = tid; i < DMODEL; i += 256) dst[i] = (src[i] - mean) * rstd * g[i] + bb[i];
    __syncthreads();
}

// ---------------------------------------------------------------------------
// One decode step for position t: embedding+PE, 2 layers (self-attn with KV
// cache, constant cross-attn add, FF), writes h[b] (rows 8..15 of hbuf stay 0)
// One block per batch element.
// ---------------------------------------------------------------------------
__global__ void step_kernel(const float* __restrict__ emb,
                            const float* __restrict__ sa_in_w,
                            const float* __restrict__ sa_in_b,
                            const float* __restrict__ sa_out_w,
                            const float* __restrict__ sa_out_b,
                            const float* __restrict__ ff1_w,
                            const float* __restrict__ ff1_b,
                            const float* __restrict__ ff2_w,
                            const float* __restrict__ ff2_b,
                            const float* __restrict__ ln1_g, const float* __restrict__ ln1_b,
                            const float* __restrict__ ln2_g, const float* __restrict__ ln2_b,
                            const float* __restrict__ ln3_g, const float* __restrict__ ln3_b,
                            const float* __restrict__ ca_add,
                            float* __restrict__ hbuf,
                            float* __restrict__ kc, float* __restrict__ vc,
                            const int* __restrict__ tok, int t) {
    __shared__ float x[DMODEL];
    __shared__ float q[DMODEL];
    __shared__ float o[DMODEL];
    __shared__ float kcur[DMODEL];
    __shared__ float vcur[DMODEL];
    __shared__ float sc[NHEAD * SMAX];
    __shared__ float u[HFF];
    __shared__ float red[256];

    int b = blockIdx.x;
    int tid = threadIdx.x;
    int token = tok[b * SMAX + t];

    // x = embedding[token] + PE[t]
    for (int i = tid; i < DMODEL; i += 256) {
        int p = i >> 1;
        float f = __expf(-(float)(2 * p) * (9.210340371976184f / (float)DMODEL));
        float ang = (float)t * f;
        x[i] = emb[(size_t)token * DMODEL + i] + ((i & 1) ? __cosf(ang) : __sinf(ang));
    }
    __syncthreads();

    for (int l = 0; l < NLAYER; l++) {
        // ---- self-attn QKV projection (1536 dots of 512) ----
        const float* W = sa_in_w + (size_t)l * 3 * DMODEL * DMODEL;
        const float* Bv = sa_in_b + (size_t)l * 3 * DMODEL;
        size_t kvbase = ((size_t)(l * BATCH + b) * SMAX + t) * DMODEL;
        for (int j = tid; j < 3 * DMODEL; j += 256) {
            float acc = Bv[j];
            const float* row = W + (size_t)j * DMODEL;
            #pragma unroll 8
            for (int k = 0; k < DMODEL; k++) acc += x[k] * row[k];
            if (j < DMODEL) {
                q[j] = acc;
            } else if (j < 2 * DMODEL) {
                kcur[j - DMODEL] = acc;
                kc[kvbase + (j - DMODEL)] = acc;
            } else {
                vcur[j - 2 * DMODEL] = acc;
                vc[kvbase + (j - 2 * DMODEL)] = acc;
            }
        }
        __syncthreads();

        // ---- scores over positions 0..t ----
        int S = t + 1;
        for (int idx = tid; idx < NHEAD * S; idx += 256) {
            int h = idx / S, p = idx % S;
            const float* kp = (p == t)
                ? (kcur + h * HDIM)
                : (kc + ((size_t)(l * BATCH + b) * SMAX + p) * DMODEL + h * HDIM);
            float acc = 0.0f;
            #pragma unroll 8
            for (int j = 0; j < HDIM; j++) acc += q[h * HDIM + j] * kp[j];
            sc[h * SMAX + p] = acc * 0.125f;  // 1/sqrt(64)
        }
        __syncthreads();

        // ---- per-head softmax (8 tiny serial softmaxes) ----
        if (tid < NHEAD) {
            int h = tid;
            float mx = -3.4e38f;
            for (int p = 0; p < S; p++) mx = fmaxf(mx, sc[h * SMAX + p]);
            float sum = 0.0f;
            for (int p = 0; p < S; p++) {
                float e = __expf(sc[h * SMAX + p] - mx);
                sc[h * SMAX + p] = e;
                sum += e;
            }
            float inv = 1.0f / sum;
            for (int p = 0; p < S; p++) sc[h * SMAX + p] *= inv;
        }
        __syncthreads();

        // ---- weighted sum of V ----
        for (int i = tid; i < DMODEL; i += 256) {
            int h = i >> 6;
            float acc = 0.0f;
            for (int p = 0; p < S; p++) {
                float vvv = (p == t)
                    ? vcur[i]
                    : vc[((size_t)(l * BATCH + b) * SMAX + p) * DMODEL + i];
                acc += sc[h * SMAX + p] * vvv;
            }
            o[i] = acc;
        }
        __syncthreads();

        // ---- output proj + residual -> q, then LN1 -> x ----
        for (int i = tid; i < DMODEL; i += 256) {
            float acc = sa_out_b[l * DMODEL + i];
            const float* row = sa_out_w + (size_t)(l * DMODEL + i) * DMODEL;
            #pragma unroll 8
            for (int k = 0; k < DMODEL; k++) acc += o[k] * row[k];
            q[i] = x[i] + acc;
        }
        __syncthreads();
        block_ln(x, q, ln1_g + l * DMODEL, ln1_b + l * DMODEL, red);

        // ---- cross-attn: constant additive vector, then LN2 ----
        for (int i = tid; i < DMODEL; i += 256)
            q[i] = x[i] + ca_add[(l * BATCH + b) * DMODEL + i];
        __syncthreads();
        block_ln(x, q, ln2_g + l * DMODEL, ln2_b + l * DMODEL, red);

        // ---- feed-forward ----
        for (int j = tid; j < HFF; j += 256) {
            float acc = ff1_b[l * HFF + j];
            const float* row = ff1_w + (size_t)(l * HFF + j) * DMODEL;
            #pragma unroll 8
            for (int k = 0; k < DMODEL; k++) acc += x[k] * row[k];
            u[j] = fmaxf(acc, 0.0f);
        }
        __syncthreads();
        for (int i = tid; i < DMODEL; i += 256) {
            float acc = ff2_b[l * DMODEL + i];
            const float* row = ff2_w + (size_t)(l * DMODEL + i) * HFF;
            #pragma unroll 8
            for (int k = 0; k < HFF; k++) acc += u[k] * row[k];
            q[i] = x[i] + acc;
        }
        __syncthreads();
        block_ln(x, q, ln3_g + l * DMODEL, ln3_b + l * DMODEL, red);
    }

    for (int i = tid; i < DMODEL; i += 256) hbuf[b * DMODEL + i] = x[i];
}

// ---------------------------------------------------------------------------
// Logits GEMM via V_WMMA_F32_16X16X4_F32:
//   logits[8x32000] = Hpad(16x512) @ out_w^T (512x32000) + out_b
// 8 waves/block, one 16-wide N tile per wave; 250 blocks cover N=32000.
// A 16x4 fragment: lane m=lane&15, K pair at k0+2*hi (hi = lane>=16).
// B 4x16 fragment: out_w row n0+(lane&15), K pair at k0+2*hi.
// D: VGPR j, lanes 0..15 hold M=j -> the 8 real batch rows.
// ---------------------------------------------------------------------------
__global__ void logits_kernel(const float* __restrict__ hbuf,
                              const float* __restrict__ outw,
                              const float* __restrict__ outb,
                              float* __restrict__ dst /* = d_out+384 + t*VOCAB */) {
    int wave = threadIdx.x >> 5;
    int lane = threadIdx.x & 31;
    int ntile = blockIdx.x * 8 + wave;
    int n0 = ntile * 16;
    int hi = (lane >> 4) & 1;
    int mn = lane & 15;

    const float* arow = hbuf + (size_t)mn * DMODEL + 2 * hi;
    const float* brow = outw + (size_t)(n0 + mn) * DMODEL + 2 * hi;

    // prefetch the weight panel for this tile (streams from L2)
    __builtin_prefetch(brow, 0, 1);
    __builtin_prefetch(brow + 128, 0, 1);
    __builtin_prefetch(brow + 256, 0, 1);
    __builtin_prefetch(brow + 384, 0, 1);

    v8f c = {0.f, 0.f, 0.f, 0.f, 0.f, 0.f, 0.f, 0.f};
    #pragma unroll 4
    for (int k0 = 0; k0 < DMODEL; k0 += 4) {
        v2f a = *(const v2f*)(arow + k0);
        v2f b = *(const v2f*)(brow + k0);
        c = __builtin_amdgcn_wmma_f32_16x16x4_f32(
            /*neg_a=*/false, a, /*neg_b=*/false, b,
            /*c_mod=*/(short)0, c, /*reuse_a=*/false, /*reuse_b=*/false);
    }

    if (lane < 16) {
        int n = n0 + lane;
        float bias = outb[n];
        #pragma unroll
        for (int j = 0; j < 8; j++)
            dst[(size_t)j * (LMAX * VOCAB) + n] = c[j] + bias;
    }
}

// ---------------------------------------------------------------------------
// Argmax over vocab per batch -> next token (first index wins ties, matching
// jnp.argmax; dividing by TEMP>0 does not change the argmax)
// ---------------------------------------------------------------------------
__global__ void argmax_kernel(const float* __restrict__ logits /* d_out+384 */,
                              int* __restrict__ tok,
                              float* __restrict__ out_tokens /* d_out */,
                              int t) {
    __shared__ float bv[256];
    __shared__ int bi[256];
    int b = blockIdx.x;
    int tid = threadIdx.x;
    const float* row = logits + (size_t)b * (LMAX * VOCAB) + (size_t)t * VOCAB;
    float best = -3.4e38f;
    int bestidx = 0x7fffffff;
    for (int n = tid; n < VOCAB; n += 256) {
        float v = row[n];
        if (v > best || (v == best && n < bestidx)) { best = v; bestidx = n; }
    }
    bv[tid] = best;
    bi[tid] = bestidx;
    __syncthreads();
    for (int s = 128; s > 0; s >>= 1) {
        if (tid < s) {
            if (bv[tid + s] > bv[tid] || (bv[tid + s] == bv[tid] && bi[tid + s] < bi[tid])) {
                bv[tid] = bv[tid + s];
                bi[tid] = bi[tid + s];
            }
        }
        __syncthreads();
    }
    if (tid == 0) {
        tok[b * SMAX + t + 1] = bi[0];
        out_tokens[b * LMAX + t] = (float)bi[0];
    }
}

// ---------------------------------------------------------------------------
extern "C" void kernel_launch(void* const* d_in, const int* in_sizes, int n_in,
                              void* d_out, int out_size, void* d_ws, size_t ws_size,
                              hipStream_t stream) {
    const float* memory    = (const float*)d_in[0];
    const float* embedding = (const float*)d_in[1];
    const float* sa_in_w   = (const float*)d_in[2];
    const float* sa_in_b   = (const float*)d_in[3];
    const float* sa_out_w  = (const float*)d_in[4];
    const float* sa_out_b  = (const float*)d_in[5];
    const float* ca_in_w   = (const float*)d_in[6];
    const float* ca_in_b   = (const float*)d_in[7];
    const float* ca_out_w  = (const float*)d_in[8];
    const float* ca_out_b  = (const float*)d_in[9];
    const float* ff1_w     = (const float*)d_in[10];
    const float* ff1_b     = (const float*)d_in[11];
    const float* ff2_w     = (const float*)d_in[12];
    const float* ff2_b     = (const float*)d_in[13];
    const float* ln1_g     = (const float*)d_in[14];
    const float* ln1_b     = (const float*)d_in[15];
    const float* ln2_g     = (const float*)d_in[16];
    const float* ln2_b     = (const float*)d_in[17];
    const float* ln3_g     = (const float*)d_in[18];
    const float* ln3_b     = (const float*)d_in[19];
    const float* out_w     = (const float*)d_in[20];
    const float* out_b     = (const float*)d_in[21];
    // d_in[22] = max_len (always 48 per setup_inputs)

    float* out = (float*)d_out;

    // workspace layout
    float* ca_add = (float*)d_ws;                                  // 2*8*512
    float* hbuf   = ca_add + NLAYER * BATCH * DMODEL;              // 16*512
    float* kc     = hbuf + 16 * DMODEL;                            // 2*8*49*512
    float* vc     = kc + (size_t)NLAYER * BATCH * SMAX * DMODEL;   // 2*8*49*512
    int*   tok    = (int*)(vc + (size_t)NLAYER * BATCH * SMAX * DMODEL);

    prep_init_kernel<<<1, 256, 0, stream>>>(hbuf, tok);
    prep_ca_kernel<<<NLAYER * BATCH, 256, 0, stream>>>(memory, ca_in_w, ca_in_b,
                                                       ca_out_w, ca_out_b, ca_add);

    float* logits_out = out + BATCH * LMAX;  // tokens (as float) occupy first 384
    for (int t = 0; t < LMAX; t++) {
        step_kernel<<<BATCH, 256, 0, stream>>>(
            embedding, sa_in_w, sa_in_b, sa_out_w, sa_out_b,
            ff1_w, ff1_b, ff2_w, ff2_b,
            ln1_g, ln1_b, ln2_g, ln2_b, ln3_g, ln3_b,
            ca_add, hbuf, kc, vc, tok, t);
        logits_kernel<<<(VOCAB / 16) / 8, 256, 0, stream>>>(
            hbuf, out_w, out_b, logits_out + (size_t)t * VOCAB);
        argmax_kernel<<<BATCH, 256, 0, stream>>>(logits_out, tok, out, t);
    }
}